// GraphMessagePassing_5952824672257
// MI455X (gfx1250) — compile-verified
//
#include <hip/hip_runtime.h>
#include <hip/hip_bf16.h>

typedef __attribute__((ext_vector_type(2))) float v2f;
typedef __attribute__((ext_vector_type(8))) float v8f;

#define D_NODE 64
#define D_EDGE 16
#define D_HID  64
#define EIN    (D_NODE + D_EDGE)   // 80
#define UIN    (D_NODE + D_HID)    // 128
// padded LDS strides (floats): even (for aligned b64) and conflict-spread mod 64
#define EIN_P  82
#define UIN_P  130
#define HID_P  66

__global__ void gmp_zero_kernel(float* __restrict__ p, int n) {
    int i = blockIdx.x * blockDim.x + threadIdx.x;
    int stride = gridDim.x * blockDim.x;
    for (; i < n; i += stride) p[i] = 0.0f;
}

// ---------------- Edge message MLP + scatter-add ----------------
// grid: ceil(E/16) blocks, 128 threads (4 waves). Each wave owns a 16-wide
// column slice of D_HID; A tile = 16 edges x 80 features staged in LDS.
__global__ void __launch_bounds__(128)
gmp_edge_kernel(const float* __restrict__ nodeF,   // [N,64]
                const float* __restrict__ edgeF,   // [E,16]
                const int*   __restrict__ eidx,    // [2,E] (src row 0, dst row 1)
                const float* __restrict__ w1,      // [80,64]
                const float* __restrict__ b1,      // [64]
                const float* __restrict__ w2,      // [64,64]
                const float* __restrict__ b2,      // [64]
                float*       __restrict__ agg,     // [N,64] accumulator
                int E) {
    __shared__ float sA[16 * EIN_P];
    __shared__ float sH[16 * HID_P];
    __shared__ int   sDst[16];

    const int tid  = threadIdx.x;
    const int lane = tid & 31;
    const int wave = tid >> 5;
    const int e0   = blockIdx.x * 16;

    // Cooperative gather: row m = [src_node_feats(64) | edge_feats(16)]
    for (int i = tid; i < 16 * EIN; i += 128) {
        const int m = i / EIN;
        const int c = i - m * EIN;
        const int e = e0 + m;
        float v = 0.0f;
        if (e < E) {
            if (c < D_NODE) {
                const int s = eidx[e];
                v = nodeF[(size_t)s * D_NODE + c];
            } else {
                v = edgeF[(size_t)e * D_EDGE + (c - D_NODE)];
            }
        }
        sA[m * EIN_P + c] = v;
    }
    if (tid < 16) {
        const int e = e0 + tid;
        sDst[tid] = (e < E) ? eidx[E + e] : -1;
    }
    __syncthreads();

    const int mrow  = lane & 15;        // A-matrix row held by this lane
    const int khalf = (lane >> 4) * 2;  // K sub-pair (0 or 2)
    const int col   = wave * 16 + mrow; // global output column of this lane

    // ---- layer 1: [16,80] @ [80,64] slice, K in steps of 4 ----
    v8f acc;
    {
        const float bias = b1[col];
        #pragma unroll
        for (int i = 0; i < 8; ++i) acc[i] = bias;
    }
    #pragma unroll
    for (int kk = 0; kk < EIN; kk += 4) {
        v2f a, b;
        a.x = sA[mrow * EIN_P + kk + khalf];
        a.y = sA[mrow * EIN_P + kk + khalf + 1];
        b.x = w1[(size_t)(kk + khalf)     * D_HID + col];
        b.y = w1[(size_t)(kk + khalf + 1) * D_HID + col];
        acc = __builtin_amdgcn_wmma_f32_16x16x4_f32(false, a, false, b,
                                                    (short)0, acc, false, false);
    }
    // ReLU, stage hidden tile to LDS (C/D layout: vgpr i -> M = i or i+8)
    #pragma unroll
    for (int i = 0; i < 8; ++i) {
        const float v = acc[i] > 0.0f ? acc[i] : 0.0f;
        const int m = (lane < 16) ? i : (i + 8);
        sH[m * HID_P + col] = v;
    }
    __syncthreads();

    // ---- layer 2: [16,64] @ [64,64] slice ----
    v8f acc2;
    {
        const float bias = b2[col];
        #pragma unroll
        for (int i = 0; i < 8; ++i) acc2[i] = bias;
    }
    #pragma unroll
    for (int kk = 0; kk < D_HID; kk += 4) {
        v2f a, b;
        a.x = sH[mrow * HID_P + kk + khalf];
        a.y = sH[mrow * HID_P + kk + khalf + 1];
        b.x = w2[(size_t)(kk + khalf)     * D_HID + col];
        b.y = w2[(size_t)(kk + khalf + 1) * D_HID + col];
        acc2 = __builtin_amdgcn_wmma_f32_16x16x4_f32(false, a, false, b,
                                                     (short)0, acc2, false, false);
    }

    // scatter-add messages into aggregation buffer (L2-resident atomics)
    #pragma unroll
    for (int i = 0; i < 8; ++i) {
        const int m = (lane < 16) ? i : (i + 8);
        const int d = sDst[m];
        if (d >= 0) atomicAdd(&agg[(size_t)d * D_HID + col], acc2[i]);
    }
}

// ---------------- Node update MLP ----------------
__global__ void __launch_bounds__(128)
gmp_node_kernel(const float* __restrict__ nodeF,   // [N,64]
                const float* __restrict__ agg,     // [N,64]
                const float* __restrict__ w1,      // [128,64]
                const float* __restrict__ b1,      // [64]
                const float* __restrict__ w2,      // [64,64]
                const float* __restrict__ b2,      // [64]
                float*       __restrict__ out,     // [N,64]
                int N) {
    __shared__ float sA[16 * UIN_P];
    __shared__ float sH[16 * HID_P];

    const int tid  = threadIdx.x;
    const int lane = tid & 31;
    const int wave = tid >> 5;
    const int n0   = blockIdx.x * 16;

    for (int i = tid; i < 16 * UIN; i += 128) {
        const int m = i >> 7;
        const int c = i & 127;
        const int node = n0 + m;
        float v = 0.0f;
        if (node < N) {
            v = (c < D_NODE) ? nodeF[(size_t)node * D_NODE + c]
                             : agg[(size_t)node * D_HID + (c - D_NODE)];
        }
        sA[m * UIN_P + c] = v;
    }
    __syncthreads();

    const int mrow  = lane & 15;
    const int khalf = (lane >> 4) * 2;
    const int col   = wave * 16 + mrow;

    v8f acc;
    {
        const float bias = b1[col];
        #pragma unroll
        for (int i = 0; i < 8; ++i) acc[i] = bias;
    }
    #pragma unroll
    for (int kk = 0; kk < UIN; kk += 4) {
        v2f a, b;
        a.x = sA[mrow * UIN_P + kk + khalf];
        a.y = sA[mrow * UIN_P + kk + khalf + 1];
        b.x = w1[(size_t)(kk + khalf)     * D_HID + col];
        b.y = w1[(size_t)(kk + khalf + 1) * D_HID + col];
        acc = __builtin_amdgcn_wmma_f32_16x16x4_f32(false, a, false, b,
                                                    (short)0, acc, false, false);
    }
    #pragma unroll
    for (int i = 0; i < 8; ++i) {
        const float v = acc[i] > 0.0f ? acc[i] : 0.0f;
        const int m = (lane < 16) ? i : (i + 8);
        sH[m * HID_P + col] = v;
    }
    __syncthreads();

    v8f acc2;
    {
        const float bias = b2[col];
        #pragma unroll
        for (int i = 0; i < 8; ++i) acc2[i] = bias;
    }
    #pragma unroll
    for (int kk = 0; kk < D_HID; kk += 4) {
        v2f a, b;
        a.x = sH[mrow * HID_P + kk + khalf];
        a.y = sH[mrow * HID_P + kk + khalf + 1];
        b.x = w2[(size_t)(kk + khalf)     * D_HID + col];
        b.y = w2[(size_t)(kk + khalf + 1) * D_HID + col];
        acc2 = __builtin_amdgcn_wmma_f32_16x16x4_f32(false, a, false, b,
                                                     (short)0, acc2, false, false);
    }

    #pragma unroll
    for (int i = 0; i < 8; ++i) {
        const int m = (lane < 16) ? i : (i + 8);
        const int node = n0 + m;
        if (node < N) out[(size_t)node * D_HID + col] = acc2[i];
    }
}

extern "C" void kernel_launch(void* const* d_in, const int* in_sizes, int n_in,
                              void* d_out, int out_size, void* d_ws, size_t ws_size,
                              hipStream_t stream) {
    const float* nodeF = (const float*)d_in[0];
    const float* edgeF = (const float*)d_in[1];
    const int*   eidx  = (const int*)  d_in[2];
    const float* w_m1  = (const float*)d_in[3];
    const float* b_m1  = (const float*)d_in[4];
    const float* w_m2  = (const float*)d_in[5];
    const float* b_m2  = (const float*)d_in[6];
    const float* w_u1  = (const float*)d_in[7];
    const float* b_u1  = (const float*)d_in[8];
    const float* w_u2  = (const float*)d_in[9];
    const float* b_u2  = (const float*)d_in[10];

    const int N = in_sizes[0] / D_NODE;       // 50000
    const int E = in_sizes[2] / 2;            // 1250000
    float* agg = (float*)d_ws;                // [N, D_HID] scratch accumulator
    float* out = (float*)d_out;               // [N, D_HID]

    gmp_zero_kernel<<<512, 256, 0, stream>>>(agg, N * D_HID);
    gmp_edge_kernel<<<(E + 15) / 16, 128, 0, stream>>>(
        nodeF, edgeF, eidx, w_m1, b_m1, w_m2, b_m2, agg, E);
    gmp_node_kernel<<<(N + 15) / 16, 128, 0, stream>>>(
        nodeF, agg, w_u1, b_u1, w_u2, b_u2, out, N);
}